// MHCLayer_61718680044185
// MI455X (gfx1250) — compile-verified
//
#include <hip/hip_runtime.h>
#include <hip/hip_bf16.h>
#include <math.h>

typedef __attribute__((ext_vector_type(2))) float v2f;
typedef __attribute__((ext_vector_type(8))) float v8f;

#define CDIM    2048
#define NBR     4
#define TOK     4                 // tokens per block (packs 16 WMMA rows)
#define ROWS    (TOK * NBR)       // 16 (token,branch) rows staged in LDS
#define LSTRIDE (CDIM + 16)       // 2064: 16n bank skew per row -> conflict-free
#define EPS     1e-5f

// ---------------------------------------------------------------------------
// Prologue: Sinkhorn on the 4x4 mixing matrix + sigmoids, once, into d_ws.
// ws[0..15] = H (row-major), ws[16..19] = pre, ws[20..23] = post
// ---------------------------------------------------------------------------
__global__ void mhc_sinkhorn_kernel(const float* __restrict__ H_res,
                                    const float* __restrict__ H_pre,
                                    const float* __restrict__ H_post,
                                    float* __restrict__ ws) {
    if (threadIdx.x != 0) return;
    float M[NBR][NBR];
    for (int i = 0; i < NBR; ++i)
        for (int j = 0; j < NBR; ++j)
            M[i][j] = expf(H_res[i * NBR + j]);
    for (int it = 0; it < 20; ++it) {
        for (int i = 0; i < NBR; ++i) {                       // row normalize
            float s = M[i][0] + M[i][1] + M[i][2] + M[i][3] + EPS;
            float inv = 1.0f / s;
            for (int j = 0; j < NBR; ++j) M[i][j] *= inv;
        }
        for (int j = 0; j < NBR; ++j) {                       // col normalize
            float s = M[0][j] + M[1][j] + M[2][j] + M[3][j] + EPS;
            float inv = 1.0f / s;
            for (int i = 0; i < NBR; ++i) M[i][j] *= inv;
        }
    }
    for (int i = 0; i < NBR; ++i)
        for (int j = 0; j < NBR; ++j)
            ws[i * NBR + j] = M[i][j];
    for (int n = 0; n < NBR; ++n) ws[16 + n] = 1.0f / (1.0f + expf(-H_pre[n]));
    for (int n = 0; n < NBR; ++n) ws[20 + n] = 1.0f / (1.0f + expf(-H_post[n]));
}

// ---------------------------------------------------------------------------
// Main fused kernel: one block = 4 tokens, single pass over x.
// Stage x slab in LDS (float4) while accumulating sum(x^2); then each wave
// emits 16-column output tiles via a 4-deep f32 WMMA chain:
//   D = sum_k A_k(16x4 slice of I4(x)H) * B_k(4x16 of x) + C(post*w*agg)
// All lane-half-dependent selects are hoisted; loop indices are constants.
// ---------------------------------------------------------------------------
__global__ void __launch_bounds__(256)
mhc_main_kernel(const float* __restrict__ x,
                const float* __restrict__ w,
                const float* __restrict__ ws,
                float* __restrict__ y) {
    extern __shared__ float xs[];          // ROWS * LSTRIDE floats (~129 KB)
    __shared__ float sums[ROWS];
    __shared__ float sArr[ROWS];

    const int tid = threadIdx.x;
    const int b0  = blockIdx.x * TOK;

    // ---- stage x into LDS + per-row sum of squares ----
    const int row    = tid >> 4;           // 0..15 : (token*4 + branch)
    const int lane16 = tid & 15;
    const float* src = x + ((size_t)(b0 * NBR + row)) * CDIM;
    float acc2 = 0.0f;
    #pragma unroll
    for (int i = 0; i < 32; ++i) {
        const int c = (lane16 + i * 16) * 4;
        const float4 v = *(const float4*)(src + c);
        *(float4*)(&xs[row * LSTRIDE + c]) = v;
        acc2 += v.x * v.x + v.y * v.y + v.z * v.z + v.w * v.w;
    }
    acc2 += __shfl_xor(acc2, 1);
    acc2 += __shfl_xor(acc2, 2);
    acc2 += __shfl_xor(acc2, 4);
    acc2 += __shfl_xor(acc2, 8);
    if (lane16 == 0) sums[row] = acc2;
    __syncthreads();
    if (tid < ROWS) sArr[tid] = rsqrtf(sums[tid] * (1.0f / (float)CDIM) + EPS);
    __syncthreads();

    // ---- lane geometry ----
    const int  lane = tid & 31;
    const bool hf   = (lane >> 4) != 0;    // lane half selects K/N pair
    const int  l16  = lane & 15;           // A: row M ; B/C/D: column
    const int  mA = l16 & 3, kA = l16 >> 2;
    const int  n0 = hf ? 2 : 0;            // B rows (== A K-indices) this half
    const int  n1 = hf ? 3 : 1;

    // ---- A chunks of I4 (x) H : built once, no runtime-indexed arrays ----
    const float hA0 = ws[mA * NBR + n0];   // per-lane L2-resident scalar loads
    const float hA1 = ws[mA * NBR + n1];
    v2f A[TOK];
    #pragma unroll
    for (int k = 0; k < TOK; ++k) {
        A[k].x = (k == kA) ? hA0 : 0.0f;
        A[k].y = (k == kA) ? hA1 : 0.0f;
    }

    // ---- hoisted pre*s products, arranged per lane half ----
    const float p0 = ws[16], p1 = ws[17], p2 = ws[18], p3 = ws[19];
    const float q0 = ws[20], q1 = ws[21], q2 = ws[22], q3 = ws[23];
    const float pre_b0 = hf ? p2 : p0;     // matches row n0
    const float pre_b1 = hf ? p3 : p1;     // matches row n1
    const float pre_o0 = hf ? p0 : p2;     // matches row n0^2
    const float pre_o1 = hf ? p1 : p3;     // matches row n1^2
    float psb0[TOK], psb1[TOK], pso0[TOK], pso1[TOK];
    #pragma unroll
    for (int k = 0; k < TOK; ++k) {
        psb0[k] = pre_b0 * sArr[k * NBR + n0];
        psb1[k] = pre_b1 * sArr[k * NBR + n1];
        pso0[k] = pre_o0 * sArr[k * NBR + (n0 ^ 2)];
        pso1[k] = pre_o1 * sArr[k * NBR + (n1 ^ 2)];
    }

    // per-lane output base: rows for this lane land at outp[r * CDIM]
    float* outp = y + ((size_t)b0 * NBR + (hf ? 8 : 0)) * CDIM;

    // uniform (scalar) tile loop
    const int wave = __builtin_amdgcn_readfirstlane(tid >> 5);
    for (int tile = wave; tile < CDIM / 16; tile += 8) {
        const int   cc  = tile * 16 + l16;
        const float wgt = w[cc];

        v2f   Bv[TOK];
        float t[TOK];
        #pragma unroll
        for (int k = 0; k < TOK; ++k) {
            const float* rowp = &xs[(k * NBR) * LSTRIDE + cc];
            const float xb0 = rowp[n0 * LSTRIDE];
            const float xb1 = rowp[n1 * LSTRIDE];
            const float xo0 = rowp[(n0 ^ 2) * LSTRIDE];
            const float xo1 = rowp[(n1 ^ 2) * LSTRIDE];
            Bv[k].x = xb0;
            Bv[k].y = xb1;
            t[k] = psb0[k] * xb0 + psb1[k] * xb1
                 + pso0[k] * xo0 + pso1[k] * xo1;
        }

        // C accumulator = post[m] * w[c] * agg_k[c]
        // D-rows: M = r + hf*8 -> m = r&3 (const), k = 2*hf + (r>=4) (2 selects)
        const float tlo = hf ? t[2] : t[0];
        const float thi = hf ? t[3] : t[1];
        const float pw0 = q0 * wgt, pw1 = q1 * wgt;
        const float pw2 = q2 * wgt, pw3 = q3 * wgt;
        v8f acc;
        acc[0] = pw0 * tlo; acc[1] = pw1 * tlo;
        acc[2] = pw2 * tlo; acc[3] = pw3 * tlo;
        acc[4] = pw0 * thi; acc[5] = pw1 * thi;
        acc[6] = pw2 * thi; acc[7] = pw3 * thi;

        // 4-deep f32 WMMA chain: D += A_k * B_k
        #pragma unroll
        for (int k = 0; k < TOK; ++k) {
            acc = __builtin_amdgcn_wmma_f32_16x16x4_f32(
                false, A[k], false, Bv[k], (short)0, acc, false, false);
        }

        // store: 8 rows at stride CDIM from the per-lane base
        #pragma unroll
        for (int r = 0; r < 8; ++r) {
            outp[(size_t)r * CDIM + cc] = acc[r];
        }
    }
}

// ---------------------------------------------------------------------------
extern "C" void kernel_launch(void* const* d_in, const int* in_sizes, int n_in,
                              void* d_out, int out_size, void* d_ws, size_t ws_size,
                              hipStream_t stream) {
    const float* x      = (const float*)d_in[0];   // [B, 4, 2048]
    const float* wrms   = (const float*)d_in[1];   // [2048]
    const float* H_pre  = (const float*)d_in[2];   // [4]
    const float* H_post = (const float*)d_in[3];   // [4]
    const float* H_res  = (const float*)d_in[4];   // [4, 4]
    float* out = (float*)d_out;
    float* ws  = (float*)d_ws;

    const int B = in_sizes[0] / (NBR * CDIM);      // 8192
    const int blocks = B / TOK;                    // 2048

    mhc_sinkhorn_kernel<<<1, 32, 0, stream>>>(H_res, H_pre, H_post, ws);

    const size_t lds = (size_t)ROWS * LSTRIDE * sizeof(float);
    mhc_main_kernel<<<blocks, 256, lds, stream>>>(x, wrms, ws, out);
}